// PointNetHead_89026082111593
// MI455X (gfx1250) — compile-verified
//
#include <hip/hip_runtime.h>

// PointNet segmentation head on MI455X (gfx1250).
//
//   msg = concat([x_j, dpos]) @ Wl  ==  (x @ Wl[:64])[src] + dpos @ Wl[64:66]
//   -> per-node fp32 WMMA GEMM (v_wmma_f32_16x16x4_f32), B staged in LDS
//   -> segment_max via ordered-uint atomicMax (L2-resident: y + agg = 33MB << 192MB L2)
//   -> agg @ Wg via the same WMMA kernel (decoding ordered-uint inline)
//   -> BN stats (column sum/sumsq) + fused normalize+ReLU.

#define NNODES 65536
#define NEDGES 1048576
#define DCH    64
#define EPSV   1e-5f

typedef __attribute__((ext_vector_type(2))) float v2f;
typedef __attribute__((ext_vector_type(8))) float v8f;

// Monotone float <-> uint mapping so unsigned atomicMax == float max.
__device__ __forceinline__ unsigned f2o(float f) {
  unsigned u = __float_as_uint(f);
  return (u >> 31) ? ~u : (u | 0x80000000u);
}
__device__ __forceinline__ float o2f(unsigned o) {
  return __uint_as_float((o >> 31) ? (o ^ 0x80000000u) : ~o);
}

// C[N x Co] = A[N x 64] @ B[64 x Co] (+bias). One wave per 16-row strip,
// 8 waves / 256 threads per block. NCT = ceil(Co/16) column tiles.
// DECODE: A holds ordered-uint segment-max results, decode inline.
//
// B is staged once per block into LDS, zero-padded to 64 x (NCT*16), with the
// (even,odd) k-pair interleaved per column so each lane's B fragment is one
// contiguous 8-byte ds_load. Row stride ≡ 32 (mod 64) dwords so the two lane
// halves (adjacent k-rows) hit disjoint LDS bank sets. The WMMA main loop is
// completely branch-free (EXEC all-ones throughout).
template <int NCT, bool DECODE>
__global__ __launch_bounds__(256) void gemm_wmma(
    const void* __restrict__ Aptr, const float* __restrict__ B,
    const float* __restrict__ bias, float* __restrict__ Cout, int Co) {
  constexpr int W      = NCT * 16;
  constexpr int STRIDE = (((2 * W) % 64) == 32) ? (2 * W) : (2 * W + 32);

  __shared__ float Bs[32 * STRIDE];

  // ---- stage B (64 x W, zero-padded) into LDS, k-pair interleaved ----
  for (int idx = threadIdx.x; idx < 64 * W; idx += 256) {
    const int k   = idx / W;
    const int col = idx - k * W;
    const float v = (col < Co) ? B[k * Co + col] : 0.0f;
    Bs[(k >> 1) * STRIDE + col * 2 + (k & 1)] = v;
  }
  __syncthreads();

  const int lane  = threadIdx.x & 31;
  const int wave  = threadIdx.x >> 5;
  const int strip = blockIdx.x * 8 + wave;
  const int m0    = strip * 16;
  const int half  = lane >> 4;   // 0: K pair {0,1}, 1: K pair {2,3}
  const int lr    = lane & 15;   // A: row-in-tile, B/C: col-in-tile
  const int arow  = m0 + lr;

  const unsigned* Au = (const unsigned*)Aptr;
  const float*    Af = (const float*)Aptr;

  v8f acc[NCT] = {};

#pragma unroll
  for (int k0 = 0; k0 < 64; k0 += 4) {
    const int ka = k0 + 2 * half;       // even
    v2f a;
    if (DECODE) {
      const uint2 u = *(const uint2*)&Au[arow * 64 + ka];
      a.x = o2f(u.x);
      a.y = o2f(u.y);
    } else {
      a = *(const v2f*)&Af[arow * 64 + ka];
    }
    const int kp = (k0 >> 1) + half;    // k-pair row in Bs
#pragma unroll
    for (int j = 0; j < NCT; ++j) {
      const v2f b = *(const v2f*)&Bs[kp * STRIDE + (j * 16 + lr) * 2];
      acc[j] = __builtin_amdgcn_wmma_f32_16x16x4_f32(
          /*neg_a=*/false, a, /*neg_b=*/false, b,
          /*c_mod=*/(short)0, acc[j], /*reuse_a=*/false, /*reuse_b=*/false);
    }
  }

#pragma unroll
  for (int j = 0; j < NCT; ++j) {
    const int col = j * 16 + lr;
    if (col < Co) {
      const float bv = bias ? bias[col] : 0.0f;
#pragma unroll
      for (int r = 0; r < 8; ++r) {
        const int row = m0 + r + 8 * half;  // C/D: VGPR r -> M=r (+8 for hi half)
        Cout[row * Co + col] = acc[j][r] + bv;
      }
    }
  }
}

// One wave per edge (incl. N trailing self-loops). Lane l handles channels 2l,2l+1.
// val = y[src][c] + dx*Wl[64][c] + dy*Wl[65][c]; ordered-uint atomicMax into agg[dst].
// y and agg are both 16MB -> resident in the 192MB L2, so the random gather and
// the atomic scatter stay on-chip.
__global__ __launch_bounds__(256) void edge_scatter(
    const float* __restrict__ y, const float* __restrict__ pos,
    const int* __restrict__ srcp, const int* __restrict__ dstp,
    const float* __restrict__ Wl,  // (66,64) row-major
    unsigned* __restrict__ agg, int nE, int total) {
  const int lane = threadIdx.x & 31;
  const int e    = (blockIdx.x * blockDim.x + threadIdx.x) >> 5;
  if (e >= total) return;

  int s, d;
  float dx = 0.0f, dy = 0.0f;
  if (e < nE) {
    s = srcp[e];
    d = dstp[e];
    const float2 ps = ((const float2*)pos)[s];
    const float2 pd = ((const float2*)pos)[d];
    dx = ps.x - pd.x;
    dy = ps.y - pd.y;
  } else {
    s = d = e - nE;  // self-loop: dpos = 0
  }

  const int c = lane * 2;
  const float2 yv = ((const float2*)(y + (size_t)s * 64))[lane];
  const float m0v = yv.x + dx * Wl[64 * 64 + c]     + dy * Wl[65 * 64 + c];
  const float m1v = yv.y + dx * Wl[64 * 64 + c + 1] + dy * Wl[65 * 64 + c + 1];
  atomicMax(&agg[(size_t)d * 64 + c],     f2o(m0v));
  atomicMax(&agg[(size_t)d * 64 + c + 1], f2o(m1v));
}

// Column sum / sum-of-squares over N rows (64 channels). 64 blocks x 1024 rows,
// 256 threads: c = tid&63 (coalesced rows), 4 row-phases per block.
__global__ __launch_bounds__(256) void bn_stats(const float* __restrict__ h,
                                                float* __restrict__ stats) {
  const int c  = threadIdx.x & 63;
  const int q  = threadIdx.x >> 6;
  const int r1 = (blockIdx.x + 1) * 1024;
  float s = 0.0f, sq = 0.0f;
  for (int r = blockIdx.x * 1024 + q; r < r1; r += 4) {
    const float v = h[(size_t)r * 64 + c];
    s += v;
    sq += v * v;
  }
  atomicAdd(&stats[c], s);
  atomicAdd(&stats[64 + c], sq);
}

// x_out = relu((h - mu) * rsqrt(var + eps) * g + b), biased var = E[h^2] - mu^2.
__global__ __launch_bounds__(256) void bn_relu(
    const float* __restrict__ h, const float* __restrict__ stats,
    const float* __restrict__ g, const float* __restrict__ b,
    float* __restrict__ xout) {
  const int i = blockIdx.x * blockDim.x + threadIdx.x;
  const int c = i & 63;
  const float invN = 1.0f / (float)NNODES;
  const float mu   = stats[c] * invN;
  const float var  = stats[64 + c] * invN - mu * mu;
  const float sc   = rsqrtf(var + EPSV) * g[c];
  const float v    = (h[i] - mu) * sc + b[c];
  xout[i] = v > 0.0f ? v : 0.0f;
}

extern "C" void kernel_launch(void* const* d_in, const int* in_sizes, int n_in,
                              void* d_out, int out_size, void* d_ws, size_t ws_size,
                              hipStream_t stream) {
  (void)in_sizes; (void)n_in; (void)out_size; (void)ws_size;

  const float* x    = (const float*)d_in[0];
  const float* pos  = (const float*)d_in[1];
  const int*   ei   = (const int*)d_in[2];
  const float* W11  = (const float*)d_in[3];
  const float* W12  = (const float*)d_in[4];
  const float* g1   = (const float*)d_in[5];
  const float* b1   = (const float*)d_in[6];
  const float* W21  = (const float*)d_in[7];
  const float* W22  = (const float*)d_in[8];
  const float* g2   = (const float*)d_in[9];
  const float* b2   = (const float*)d_in[10];
  const float* W31  = (const float*)d_in[11];
  const float* W32  = (const float*)d_in[12];
  const float* g3   = (const float*)d_in[13];
  const float* b3   = (const float*)d_in[14];
  const float* Wr1  = (const float*)d_in[15];
  const float* Wr2  = (const float*)d_in[16];
  const float* br2  = (const float*)d_in[17];
  const float* Wc1  = (const float*)d_in[18];
  const float* Wc2  = (const float*)d_in[19];
  const float* bc2  = (const float*)d_in[20];
  const float* Wo1  = (const float*)d_in[21];
  const float* Wo2  = (const float*)d_in[22];
  const float* bo2  = (const float*)d_in[23];

  const int* srcp = ei;
  const int* dstp = ei + NEDGES;

  const size_t SZ = (size_t)NNODES * DCH * sizeof(float);  // 16 MiB
  char* ws = (char*)d_ws;
  float*    ybuf  = (float*)(ws + 0 * SZ);   // y, then reused as h
  unsigned* agg   = (unsigned*)(ws + 1 * SZ);
  float*    x1    = (float*)(ws + 2 * SZ);
  float*    x2    = (float*)(ws + 3 * SZ);
  float*    x3    = (float*)(ws + 4 * SZ);
  float*    stats = (float*)(ws + 5 * SZ);   // 128 floats

  float* out = (float*)d_out;
  float* cls = out;                              // [N,101]
  float* reg = out + (size_t)NNODES * 101;       // [N,4]
  float* obj = out + (size_t)NNODES * 105;       // [N,1]

  const int GEMM_GRID = NNODES / 16 / 8;          // 512 blocks of 8 waves, exact
  const int EBLK      = (NEDGES + NNODES) / 8;    // 8 edges per 256-thread block

  auto conv_front = [&](const float* xin, const float* Wl) {
    // y = xin @ Wl[:64]   (WMMA fp32)
    gemm_wmma<4, false><<<GEMM_GRID, 256, 0, stream>>>(xin, Wl, nullptr, ybuf, 64);
    // agg = below-everything in ordered space
    hipMemsetAsync(agg, 0, SZ, stream);
    // segment-max over edges + self-loops
    edge_scatter<<<EBLK, 256, 0, stream>>>(ybuf, pos, srcp, dstp, Wl, agg,
                                           NEDGES, NEDGES + NNODES);
  };

  auto conv64 = [&](const float* xin, const float* Wl, const float* Wg,
                    const float* gg, const float* bb, float* xout) {
    conv_front(xin, Wl);
    // h = decode(agg) @ Wg   (reuses ybuf)
    gemm_wmma<4, true><<<GEMM_GRID, 256, 0, stream>>>(agg, Wg, nullptr, ybuf, 64);
    hipMemsetAsync(stats, 0, 128 * sizeof(float), stream);
    bn_stats<<<64, 256, 0, stream>>>(ybuf, stats);
    bn_relu<<<(NNODES * DCH) / 256, 256, 0, stream>>>(ybuf, stats, gg, bb, xout);
  };

  auto head = [&](const float* xin, const float* Wl, const float* Wg,
                  const float* bias, float* outp, int Co) {
    conv_front(xin, Wl);
    if (Co > 16)
      gemm_wmma<7, true><<<GEMM_GRID, 256, 0, stream>>>(agg, Wg, bias, outp, Co);
    else
      gemm_wmma<1, true><<<GEMM_GRID, 256, 0, stream>>>(agg, Wg, bias, outp, Co);
  };

  conv64(x,  W11, W12, g1, b1, x1);
  conv64(x1, W21, W22, g2, b2, x2);
  conv64(x1, W31, W32, g3, b3, x3);
  head(x2, Wr1, Wr2, br2, reg, 4);
  head(x3, Wc1, Wc2, bc2, cls, 101);
  head(x3, Wo1, Wo2, bo2, obj, 1);
}